// GAT_11089605558973
// MI455X (gfx1250) — compile-verified
//
#include <hip/hip_runtime.h>
#include <hip/hip_bf16.h>

#define N_NODES 50000
#define N_EDGES 800000
#define DIN     256
#define HID     32
#define HEADS   8
#define DH      256   /* HEADS*HID */
#define DOUT    40
#define SLOPE   0.2f

typedef __attribute__((ext_vector_type(2))) float v2f;
typedef __attribute__((ext_vector_type(8))) float v8f;

// ---------------------------------------------------------------------------
// GEMM1: G1[N,256] = X[N,256] @ W1[256,256], fp32 WMMA 16x16x4.
// One wave per 16-row x 64-col tile (4 accumulators). grid = (4, N/16), block=32.
// ---------------------------------------------------------------------------
__global__ __launch_bounds__(32) void gat_gemm1_wmma(
    const float* __restrict__ X, const float* __restrict__ W1,
    float* __restrict__ G1) {
  const int lane = threadIdx.x;
  const int half = lane >> 4;        // 0: K even pair, 1: K odd pair
  const int l    = lane & 15;
  const int rowTile = blockIdx.y;
  const int nbase   = blockIdx.x * 64;
  const int m = rowTile * 16 + l;
  const float* __restrict__ xrow = X + (size_t)m * DIN;

  v8f c0 = {}, c1 = {}, c2 = {}, c3 = {};
  for (int k = 0; k < DIN; k += 4) {
    const int kk = k + 2 * half;
    v2f a;
    a.x = xrow[kk];
    a.y = xrow[kk + 1];
    const float* __restrict__ wp = W1 + (size_t)kk * DH + nbase + l;
    v2f b0, b1, b2, b3;
    b0.x = wp[0];        b0.y = wp[DH];
    b1.x = wp[16];       b1.y = wp[16 + DH];
    b2.x = wp[32];       b2.y = wp[32 + DH];
    b3.x = wp[48];       b3.y = wp[48 + DH];
    c0 = __builtin_amdgcn_wmma_f32_16x16x4_f32(false, a, false, b0, (short)0, c0, false, false);
    c1 = __builtin_amdgcn_wmma_f32_16x16x4_f32(false, a, false, b1, (short)0, c1, false, false);
    c2 = __builtin_amdgcn_wmma_f32_16x16x4_f32(false, a, false, b2, (short)0, c2, false, false);
    c3 = __builtin_amdgcn_wmma_f32_16x16x4_f32(false, a, false, b3, (short)0, c3, false, false);
  }
  for (int r = 0; r < 8; ++r) {
    const int row = rowTile * 16 + r + 8 * half;
    float* out = G1 + (size_t)row * DH + nbase + l;
    out[0]  = c0[r];
    out[16] = c1[r];
    out[32] = c2[r];
    out[48] = c3[r];
  }
}

// ---------------------------------------------------------------------------
// GEMM2: G2[N,40] = H[N,256] @ W2[256,40], fp32 WMMA 16x16x4.
// grid = (3, N/16) (cols 0..47, guarded to <40), block=32.
// ---------------------------------------------------------------------------
__global__ __launch_bounds__(32) void gat_gemm2_wmma(
    const float* __restrict__ H, const float* __restrict__ W2,
    float* __restrict__ G2) {
  const int lane = threadIdx.x;
  const int half = lane >> 4;
  const int l    = lane & 15;
  const int rowTile = blockIdx.y;
  const int n = blockIdx.x * 16 + l;
  const int m = rowTile * 16 + l;
  const float* __restrict__ hrow = H + (size_t)m * DH;
  const bool nok = (n < DOUT);

  v8f c = {};
  for (int k = 0; k < DH; k += 4) {
    const int kk = k + 2 * half;
    v2f a, b;
    a.x = hrow[kk];
    a.y = hrow[kk + 1];
    b.x = nok ? W2[(size_t)kk * DOUT + n]       : 0.0f;
    b.y = nok ? W2[(size_t)(kk + 1) * DOUT + n] : 0.0f;
    c = __builtin_amdgcn_wmma_f32_16x16x4_f32(false, a, false, b, (short)0, c, false, false);
  }
  if (nok) {
    for (int r = 0; r < 8; ++r) {
      const int row = rowTile * 16 + r + 8 * half;
      G2[(size_t)row * DOUT + n] = c[r];
    }
  }
}

// ---------------------------------------------------------------------------
// Per-node attention coefficients, layer 1: as1/ad1[n,h] = <G1[n,h,:], a1s/a1d[h,:]>
// ---------------------------------------------------------------------------
__global__ void gat_alpha1(const float* __restrict__ G1,
                           const float* __restrict__ a1s, const float* __restrict__ a1d,
                           float* __restrict__ as1, float* __restrict__ ad1) {
  const int idx = blockIdx.x * blockDim.x + threadIdx.x;
  if (idx >= N_NODES * HEADS) return;
  const int n = idx >> 3, h = idx & 7;
  const float* g = G1 + (size_t)n * DH + h * HID;
  const float* s = a1s + h * HID;
  const float* d = a1d + h * HID;
  float ss = 0.f, dd = 0.f;
  for (int f = 0; f < HID; ++f) { ss += g[f] * s[f]; dd += g[f] * d[f]; }
  as1[idx] = ss;
  ad1[idx] = dd;
}

__global__ void gat_alpha2(const float* __restrict__ G2,
                           const float* __restrict__ a2s, const float* __restrict__ a2d,
                           float* __restrict__ as2, float* __restrict__ ad2) {
  const int n = blockIdx.x * blockDim.x + threadIdx.x;
  if (n >= N_NODES) return;
  const float* g = G2 + (size_t)n * DOUT;
  float ss = 0.f, dd = 0.f;
  for (int f = 0; f < DOUT; ++f) { ss += g[f] * a2s[f]; dd += g[f] * a2d[f]; }
  as2[n] = ss;
  ad2[n] = dd;
}

__global__ void gat_fill(float* __restrict__ p, int n, float v) {
  const int i = blockIdx.x * blockDim.x + threadIdx.x;
  if (i < n) p[i] = v;
}

// ---------------------------------------------------------------------------
// Layer-1 edge passes (8 heads)
// ---------------------------------------------------------------------------
__global__ void gat_edge1_score(const int* __restrict__ src, const int* __restrict__ dst,
                                const float* __restrict__ as1, const float* __restrict__ ad1,
                                float* __restrict__ EB, float* __restrict__ m1) {
  const int idx = blockIdx.x * blockDim.x + threadIdx.x;
  if (idx >= N_EDGES * HEADS) return;
  const int e = idx >> 3, h = idx & 7;
  const int s = src[e], d = dst[e];
  float v = as1[s * HEADS + h] + ad1[d * HEADS + h];
  v = v > 0.f ? v : SLOPE * v;
  EB[idx] = v;
  atomicMax(&m1[d * HEADS + h], v);
}

__global__ void gat_edge1_exp(const int* __restrict__ dst,
                              const float* __restrict__ m1, float* __restrict__ EB,
                              float* __restrict__ dn1) {
  const int idx = blockIdx.x * blockDim.x + threadIdx.x;
  if (idx >= N_EDGES * HEADS) return;
  const int e = idx >> 3, h = idx & 7;
  const int d = dst[e];
  const float ex = expf(EB[idx] - m1[d * HEADS + h]);
  EB[idx] = ex;
  atomicAdd(&dn1[d * HEADS + h], ex);
}

__global__ void gat_edge1_msg(const int* __restrict__ src, const int* __restrict__ dst,
                              const float* __restrict__ Wd,
                              const float* __restrict__ G1, const float* __restrict__ EB,
                              const float* __restrict__ dn1, float* __restrict__ AGG1) {
  const long long idx = (long long)blockIdx.x * blockDim.x + threadIdx.x;  // E*256 threads
  const int e = (int)(idx >> 8);
  const int c = (int)(idx & 255);
  const int h = c >> 5;
  const int s = src[e], d = dst[e];
  const float alpha = EB[(size_t)e * HEADS + h] / (dn1[d * HEADS + h] + 1e-16f);
  const float val = G1[(size_t)s * DH + c] * alpha * Wd[e];
  atomicAdd(&AGG1[(size_t)d * DH + c], val);
}

__global__ void gat_relu_bias(float* __restrict__ AGG1, const float* __restrict__ b1) {
  const long long idx = (long long)blockIdx.x * blockDim.x + threadIdx.x;  // N*256
  const float v = AGG1[idx] + b1[idx & 255];
  AGG1[idx] = v > 0.f ? v : 0.f;
}

// ---------------------------------------------------------------------------
// Layer-2 edge passes (1 head, 40 features)
// ---------------------------------------------------------------------------
__global__ void gat_edge2_score(const int* __restrict__ src, const int* __restrict__ dst,
                                const float* __restrict__ as2, const float* __restrict__ ad2,
                                float* __restrict__ EB, float* __restrict__ m2) {
  const int e = blockIdx.x * blockDim.x + threadIdx.x;
  if (e >= N_EDGES) return;
  float v = as2[src[e]] + ad2[dst[e]];
  v = v > 0.f ? v : SLOPE * v;
  EB[e] = v;
  atomicMax(&m2[dst[e]], v);
}

__global__ void gat_edge2_exp(const int* __restrict__ dst,
                              const float* __restrict__ m2, float* __restrict__ EB,
                              float* __restrict__ dn2) {
  const int e = blockIdx.x * blockDim.x + threadIdx.x;
  if (e >= N_EDGES) return;
  const float ex = expf(EB[e] - m2[dst[e]]);
  EB[e] = ex;
  atomicAdd(&dn2[dst[e]], ex);
}

__global__ void gat_edge2_msg(const int* __restrict__ src, const int* __restrict__ dst,
                              const float* __restrict__ Wd,
                              const float* __restrict__ G2, const float* __restrict__ EB,
                              const float* __restrict__ dn2, float* __restrict__ AGG2) {
  const long long idx = (long long)blockIdx.x * blockDim.x + threadIdx.x;  // E*64 threads
  const int e = (int)(idx >> 6);
  const int c = (int)(idx & 63);
  if (c >= DOUT) return;
  const int s = src[e], d = dst[e];
  const float alpha = EB[e] / (dn2[d] + 1e-16f);
  const float val = G2[(size_t)s * DOUT + c] * alpha * Wd[e];
  atomicAdd(&AGG2[(size_t)d * DOUT + c], val);
}

__global__ void gat_logsoftmax(const float* __restrict__ AGG2, const float* __restrict__ b2,
                               float* __restrict__ out) {
  const int n = blockIdx.x * blockDim.x + threadIdx.x;
  if (n >= N_NODES) return;
  float z[DOUT];
  float mx = -3.402823466e+38f;
  for (int f = 0; f < DOUT; ++f) {
    z[f] = AGG2[(size_t)n * DOUT + f] + b2[f];
    mx = z[f] > mx ? z[f] : mx;
  }
  float s = 0.f;
  for (int f = 0; f < DOUT; ++f) s += expf(z[f] - mx);
  const float ls = logf(s) + mx;
  for (int f = 0; f < DOUT; ++f) out[(size_t)n * DOUT + f] = z[f] - ls;
}

// ---------------------------------------------------------------------------
extern "C" void kernel_launch(void* const* d_in, const int* in_sizes, int n_in,
                              void* d_out, int out_size, void* d_ws, size_t ws_size,
                              hipStream_t stream) {
  const float* X   = (const float*)d_in[0];
  const int*   A   = (const int*)d_in[1];
  const float* Wd  = (const float*)d_in[2];
  const float* W1  = (const float*)d_in[3];
  const float* a1s = (const float*)d_in[4];
  const float* a1d = (const float*)d_in[5];
  const float* b1  = (const float*)d_in[6];
  const float* W2  = (const float*)d_in[7];
  const float* a2s = (const float*)d_in[8];
  const float* a2d = (const float*)d_in[9];
  const float* b2  = (const float*)d_in[10];
  float* out = (float*)d_out;

  const int* src = A;
  const int* dst = A + N_EDGES;

  // Workspace layout (floats)
  float* ws = (float*)d_ws;
  size_t off = 0;
  float* G1   = ws + off; off += (size_t)N_NODES * DH;      // 12.8M
  float* AGG1 = ws + off; off += (size_t)N_NODES * DH;      // 12.8M (becomes H after relu)
  float* EB   = ws + off; off += (size_t)N_EDGES * HEADS;   // 6.4M
  float* as1  = ws + off; off += (size_t)N_NODES * HEADS;
  float* ad1  = ws + off; off += (size_t)N_NODES * HEADS;
  float* m1   = ws + off; off += (size_t)N_NODES * HEADS;
  float* dn1  = ws + off; off += (size_t)N_NODES * HEADS;
  float* G2   = ws + off; off += (size_t)N_NODES * DOUT;
  float* AGG2 = ws + off; off += (size_t)N_NODES * DOUT;
  float* as2  = ws + off; off += (size_t)N_NODES;
  float* ad2  = ws + off; off += (size_t)N_NODES;
  float* m2   = ws + off; off += (size_t)N_NODES;
  float* dn2  = ws + off; off += (size_t)N_NODES;
  (void)off; (void)ws_size; (void)in_sizes; (void)n_in; (void)out_size;

  // Zero-init accumulators (graph-capturable async memsets)
  hipMemsetAsync(AGG1, 0, (size_t)N_NODES * DH * sizeof(float), stream);
  hipMemsetAsync(dn1,  0, (size_t)N_NODES * HEADS * sizeof(float), stream);
  hipMemsetAsync(AGG2, 0, (size_t)N_NODES * DOUT * sizeof(float), stream);
  hipMemsetAsync(dn2,  0, (size_t)N_NODES * sizeof(float), stream);
  gat_fill<<<(N_NODES * HEADS + 255) / 256, 256, 0, stream>>>(m1, N_NODES * HEADS, -INFINITY);
  gat_fill<<<(N_NODES + 255) / 256, 256, 0, stream>>>(m2, N_NODES, -INFINITY);

  // ---- Layer 1 ----
  gat_gemm1_wmma<<<dim3(4, N_NODES / 16), 32, 0, stream>>>(X, W1, G1);
  gat_alpha1<<<(N_NODES * HEADS + 255) / 256, 256, 0, stream>>>(G1, a1s, a1d, as1, ad1);
  gat_edge1_score<<<(N_EDGES * HEADS) / 256, 256, 0, stream>>>(src, dst, as1, ad1, EB, m1);
  gat_edge1_exp<<<(N_EDGES * HEADS) / 256, 256, 0, stream>>>(dst, m1, EB, dn1);
  gat_edge1_msg<<<N_EDGES, 256, 0, stream>>>(src, dst, Wd, G1, EB, dn1, AGG1);
  gat_relu_bias<<<N_NODES * DH / 256, 256, 0, stream>>>(AGG1, b1);

  // ---- Layer 2 ----
  gat_gemm2_wmma<<<dim3(3, N_NODES / 16), 32, 0, stream>>>(AGG1, W2, G2);
  gat_alpha2<<<(N_NODES + 255) / 256, 256, 0, stream>>>(G2, a2s, a2d, as2, ad2);
  gat_edge2_score<<<(N_EDGES + 255) / 256, 256, 0, stream>>>(src, dst, as2, ad2, EB, m2);
  gat_edge2_exp<<<(N_EDGES + 255) / 256, 256, 0, stream>>>(dst, m2, EB, dn2);
  gat_edge2_msg<<<(N_EDGES * 64) / 256, 256, 0, stream>>>(src, dst, Wd, G2, EB, dn2, AGG2);

  // ---- log_softmax ----
  gat_logsoftmax<<<(N_NODES + 255) / 256, 256, 0, stream>>>(AGG2, b2, out);
}